// Fitness_29918742183959
// MI455X (gfx1250) — compile-verified
//
#include <hip/hip_runtime.h>
#include <cstdint>

// Problem shape (reference: B=2048 rows, C=128000 cols, f32).
#define C_COLS        128000
#define THREADS       256
#define WAVES         8           // 256 threads / wave32
#define F4_PER_THREAD 5
#define LANE_STRIDE   128         // floats between a lane's consecutive float4 slots
#define WAVE_SLICE    640         // floats per wave per chunk  (5 * 128)
#define CHUNK         5120        // floats per chunk = THREADS*4*F4_PER_THREAD
#define NCHUNK        25          // 25 * 5120 = 128000 exactly

// ---- CDNA5 async global->LDS copies (ASYNCcnt path) ------------------------
// ISA: each b128 op copies 16 bytes per lane from global memory into LDS;
// the immediate offset advances BOTH sides, which matches our chunk layout,
// so one 64-bit global base + one 32-bit LDS base covers all 5 issues.
__device__ __forceinline__ void issue_chunk(uint32_t lds_base, unsigned long long g) {
    asm volatile("global_load_async_to_lds_b128 %0, %1, off"
                 :: "v"(lds_base), "v"(g) : "memory");
    asm volatile("global_load_async_to_lds_b128 %0, %1, off offset:512"
                 :: "v"(lds_base), "v"(g) : "memory");
    asm volatile("global_load_async_to_lds_b128 %0, %1, off offset:1024"
                 :: "v"(lds_base), "v"(g) : "memory");
    asm volatile("global_load_async_to_lds_b128 %0, %1, off offset:1536"
                 :: "v"(lds_base), "v"(g) : "memory");
    asm volatile("global_load_async_to_lds_b128 %0, %1, off offset:2048"
                 :: "v"(lds_base), "v"(g) : "memory");
}
__device__ __forceinline__ void wait_async_5() {
    asm volatile("s_wait_asynccnt 5" ::: "memory");
}
__device__ __forceinline__ void wait_async_0() {
    asm volatile("s_wait_asynccnt 0" ::: "memory");
}

// Merge another (v1,i1,v2) candidate set into ours (branchless; ties -> lower
// index, matching jax.lax.top_k stability). Second-best index is never needed.
__device__ __forceinline__ void merge_top2(float& v1, int& i1, float& v2,
                                           float ov1, int oi1, float ov2) {
    const bool og   = (ov1 > v1) || ((ov1 == v1) && (oi1 < i1));
    const float los = og ? v1 : ov1;              // loser of the max battle
    v2 = fmaxf(fmaxf(v2, ov2), los);
    v1 = og ? ov1 : v1;
    i1 = og ? oi1 : i1;
}

__global__ __launch_bounds__(THREADS, 1)
void Fitness_ce_margin_kernel(const float* __restrict__ logits,
                              const int* __restrict__ y,
                              float* __restrict__ out) {
    __shared__ float buf[2][CHUNK];
    __shared__ float s_v1[WAVES], s_v2[WAVES], s_sum[WAVES];
    __shared__ int   s_i1[WAVES];

    const int row  = blockIdx.x;
    const int tid  = threadIdx.x;
    const int wave = tid >> 5;
    const int lane = tid & 31;
    // This lane's base float offset inside a chunk (j-th slot adds j*LANE_STRIDE).
    const int slot = wave * WAVE_SLICE + lane * 4;

    const float* rowp = logits + (size_t)row * (size_t)C_COLS;

    // Raw LDS byte addresses (flat-LDS low 32 bits == LDS byte offset).
    const uint32_t lds0 = (uint32_t)(uintptr_t)&buf[0][slot];
    const uint32_t lds1 = (uint32_t)(uintptr_t)&buf[1][slot];

    // ---- prologue: prefetch chunk 0 into buffer 0 ----
    unsigned long long g = (unsigned long long)(rowp + slot);
    issue_chunk(lds0, g);

    float v1 = -__builtin_inff(), v2 = -__builtin_inff();
    int   i1 = 0x7fffffff;
    float s0 = 0.f, s1 = 0.f, s2 = 0.f, s3 = 0.f;

    for (int c = 0; c < NCHUNK; ++c) {
        // Prefetch next chunk into the other buffer, then wait for the current
        // chunk (its 5 ops are the oldest; async loads complete in order).
        if (c + 1 < NCHUNK) {
            g += (unsigned long long)(CHUNK * sizeof(float));
            issue_chunk(((c + 1) & 1) ? lds1 : lds0, g);
            wait_async_5();
        } else {
            wait_async_0();
        }

        const float* lbuf = (c & 1) ? buf[1] : buf[0];
        const int colbase = c * CHUNK + slot;
#pragma unroll
        for (int j = 0; j < F4_PER_THREAD; ++j) {
            const float4 q = *(const float4*)&lbuf[slot + j * LANE_STRIDE];
            const int cb = colbase + j * LANE_STRIDE;
            const float xs[4] = {q.x, q.y, q.z, q.w};
            // Branchless running top-2 (values) + argmax index; 4 independent
            // exp-sum chains. All select logic is v_cndmask/v_max (no exec churn).
            {
                const float x = xs[0]; s0 += __expf(x);
                const bool gtb = x > v1;
                v2 = fmaxf(v2, fminf(x, v1));
                i1 = gtb ? (cb + 0) : i1;
                v1 = fmaxf(v1, x);
            }
            {
                const float x = xs[1]; s1 += __expf(x);
                const bool gtb = x > v1;
                v2 = fmaxf(v2, fminf(x, v1));
                i1 = gtb ? (cb + 1) : i1;
                v1 = fmaxf(v1, x);
            }
            {
                const float x = xs[2]; s2 += __expf(x);
                const bool gtb = x > v1;
                v2 = fmaxf(v2, fminf(x, v1));
                i1 = gtb ? (cb + 2) : i1;
                v1 = fmaxf(v1, x);
            }
            {
                const float x = xs[3]; s3 += __expf(x);
                const bool gtb = x > v1;
                v2 = fmaxf(v2, fminf(x, v1));
                i1 = gtb ? (cb + 3) : i1;
                v1 = fmaxf(v1, x);
            }
        }
    }

    float sum = (s0 + s1) + (s2 + s3);

    // ---- wave32 butterfly reduction of (v1,i1,v2,sum) ----
#pragma unroll
    for (int m = 16; m >= 1; m >>= 1) {
        const float ov1 = __shfl_xor(v1, m);
        const int   oi1 = __shfl_xor(i1, m);
        const float ov2 = __shfl_xor(v2, m);
        const float os  = __shfl_xor(sum, m);
        sum += os;
        merge_top2(v1, i1, v2, ov1, oi1, ov2);
    }

    if (lane == 0) {
        s_v1[wave] = v1; s_i1[wave] = i1;
        s_v2[wave] = v2; s_sum[wave] = sum;
    }
    __syncthreads();

    if (tid == 0) {
        float V1 = s_v1[0], V2 = s_v2[0], S = s_sum[0];
        int   I1 = s_i1[0];
#pragma unroll
        for (int w = 1; w < WAVES; ++w) {
            S += s_sum[w];
            merge_top2(V1, I1, V2, s_v1[w], s_i1[w], s_v2[w]);
        }
        const int yy = y[row];
        const float first = (I1 == yy) ? V2 : V1;
        // Faithful to reference: subtract the raw logit from sum(exp), then log.
        out[row] = first - __logf(S - first);
    }
}

extern "C" void kernel_launch(void* const* d_in, const int* in_sizes, int n_in,
                              void* d_out, int out_size, void* d_ws, size_t ws_size,
                              hipStream_t stream) {
    (void)in_sizes; (void)n_in; (void)d_ws; (void)ws_size;
    const float* logits = (const float*)d_in[0];
    const int*   y      = (const int*)d_in[1];
    float*       out    = (float*)d_out;

    dim3 grid((unsigned)out_size);   // one block per row (2048)
    dim3 block(THREADS);
    hipLaunchKernelGGL(Fitness_ce_margin_kernel, grid, block, 0, stream,
                       logits, y, out);
}